// GNNWithEmbeddings_83176336654747
// MI455X (gfx1250) — compile-verified
//
#include <hip/hip_runtime.h>

// ---------------------------------------------------------------------------
// GCN 2-layer forward for gfx1250 (MI455X). Edge scatter is HBM-bound
// (~4 GB total traffic -> ~170us at 23.3 TB/s); GEMMs run on the f32 WMMA
// path (V_WMMA_F32_16X16X4_F32) since they are tiny (2.4 GFLOP total) and
// inputs are fp32.
// ---------------------------------------------------------------------------

typedef __attribute__((ext_vector_type(2))) float v2f;
typedef __attribute__((ext_vector_type(8))) float v8f;

__device__ __forceinline__ void gatomic_add_f32(float* p, float v) {
    // Non-returning f32 add at device scope (goes to L2 atomic units).
    asm volatile("global_atomic_add_f32 %0, %1, off scope:SCOPE_DEV"
                 :: "v"(p), "v"(v) : "memory");
}

// ---------------- degree / norm ----------------
__global__ void deg_init(unsigned* deg, int n) {
    int i = blockIdx.x * blockDim.x + threadIdx.x;
    if (i < n) deg[i] = 1u;  // self loop
}

__global__ void deg_count(const int* __restrict__ dst, unsigned* deg, int e) {
    int i = blockIdx.x * blockDim.x + threadIdx.x;
    if (i < e) atomicAdd(&deg[dst[i]], 1u);
}

__global__ void deg_finalize(float* dinv, int n) {
    int i = blockIdx.x * blockDim.x + threadIdx.x;
    if (i < n) {
        unsigned d = ((const unsigned*)dinv)[i];
        dinv[i] = rsqrtf((float)d);   // d >= 1 always
    }
}

// ---------------- GEMM: Y[N x 64] = X[N x K] @ W[K x 64] ----------------
// 256 threads = 8 waves; wave (r,c) computes the 16x16 tile at
// (blockIdx.x*32 + r*16, c*16). K stepped by 4 through V_WMMA_F32_16X16X4_F32.
// A frag (16x4 f32): lanes 0-15 row M=l hold K={k0,k0+1}; lanes 16-31 hold
// K={k0+2,k0+3}. B frag (4x16) mirrors along K, N striped across lanes.
__global__ void gemm_wmma_f32(const float* __restrict__ X, int K,
                              const float* __restrict__ W,
                              float* __restrict__ Y, int N) {
    const int lane = threadIdx.x & 31;
    const int wave = threadIdx.x >> 5;
    const int half = lane >> 4;          // 0 or 1
    const int l    = lane & 15;
    const int colBase = (wave & 3) * 16; // 4 tiles cover 64 cols
    const int rowBase = blockIdx.x * 32 + (wave >> 2) * 16;
    if (rowBase >= N) return;            // uniform per wave (N % 16 == 0)

    v8f acc = {};
    const float* xrow = X + (size_t)(rowBase + l) * K;
    for (int k0 = 0; k0 < K; k0 += 4) {
        v2f a, b;
        const int ka = k0 + 2 * half;
        a.x = xrow[ka];
        a.y = xrow[ka + 1];
        const float* wr = W + (size_t)ka * 64 + colBase + l;
        b.x = wr[0];
        b.y = wr[64];
        acc = __builtin_amdgcn_wmma_f32_16x16x4_f32(
            /*neg_a=*/false, a, /*neg_b=*/false, b,
            /*c_mod=*/(short)0, acc, /*reuse_a=*/false, /*reuse_b=*/false);
    }
    // C/D layout: VGPR v -> row (v + half*8), col l of the tile.
    float* yout = Y + (size_t)(rowBase + half * 8) * 64 + colBase + l;
#pragma unroll
    for (int v = 0; v < 8; ++v)
        yout[(size_t)v * 64] = acc[v];
}

// ---------------- self-loop init: acc[i,:] = y[i,:] * dinv[i]^2 ----------
__global__ void selfloop_init(const float* __restrict__ y,
                              const float* __restrict__ dinv,
                              float* __restrict__ acc, size_t total) {
    size_t gid = (size_t)blockIdx.x * blockDim.x + threadIdx.x;
    if (gid < total) {
        float di = dinv[gid >> 6];
        acc[gid] = y[gid] * di * di;
    }
}

// ---------------- edge scatter: one wave32 per edge ----------------------
// Each lane handles 2 contiguous features -> 256B coalesced row read,
// two f32 global atomics per lane into the destination row.
__global__ void edge_scatter(const int* __restrict__ src,
                             const int* __restrict__ dst,
                             const float* __restrict__ dinv,
                             const float* __restrict__ y,
                             float* acc, int E) {
    size_t gid = (size_t)blockIdx.x * blockDim.x + threadIdx.x;
    int e = (int)(gid >> 5);
    int lane = (int)(gid & 31);
    if (e >= E) return;
    int s = src[e];
    int d = dst[e];
    float norm = dinv[s] * dinv[d];
    float2 v = ((const float2*)(y + (size_t)s * 64))[lane];
    float* pd = acc + (size_t)d * 64 + lane * 2;
    gatomic_add_f32(pd,     v.x * norm);
    gatomic_add_f32(pd + 1, v.y * norm);
}

// ---------------- elementwise epilogues ----------------------------------
__global__ void bias_relu_inplace(float* h, const float* __restrict__ b, size_t total) {
    size_t gid = (size_t)blockIdx.x * blockDim.x + threadIdx.x;
    if (gid < total) h[gid] = fmaxf(h[gid] + b[gid & 63], 0.0f);
}

__global__ void bias_add_inplace(float* o, const float* __restrict__ b, size_t total) {
    size_t gid = (size_t)blockIdx.x * blockDim.x + threadIdx.x;
    if (gid < total) o[gid] += b[gid & 63];
}

// ---------------------------------------------------------------------------
extern "C" void kernel_launch(void* const* d_in, const int* in_sizes, int n_in,
                              void* d_out, int out_size, void* d_ws, size_t ws_size,
                              hipStream_t stream) {
    const float* emb = (const float*)d_in[0];
    const float* W1  = (const float*)d_in[1];
    const float* b1  = (const float*)d_in[2];
    const float* W2  = (const float*)d_in[3];
    const float* b2  = (const float*)d_in[4];
    const int*   src = (const int*)d_in[5];
    const int*   dst = (const int*)d_in[6];

    const int H = in_sizes[2];           // 64
    const int D = in_sizes[1] / H;       // 128
    const int N = in_sizes[0] / D;       // 100000
    const int E = in_sizes[5];           // 1600000
    const size_t NF = (size_t)N * 64;

    float* dinv = (float*)d_ws;          // N floats (u32 degree first, then dinv)
    float* y    = dinv + N;              // N*64: linear-transform output
    float* h    = y + NF;                // N*64: aggregation accumulator / hidden
    float* out  = (float*)d_out;         // N*64

    const int T = 256;
    dim3 blk(T);

    // degree + symmetric norm
    deg_init    <<<(N + T - 1) / T, blk, 0, stream>>>((unsigned*)dinv, N);
    deg_count   <<<(E + T - 1) / T, blk, 0, stream>>>(dst, (unsigned*)dinv, E);
    deg_finalize<<<(N + T - 1) / T, blk, 0, stream>>>(dinv, N);

    // ---- layer 1 ----
    gemm_wmma_f32<<<(N + 31) / 32, blk, 0, stream>>>(emb, D, W1, y, N);
    selfloop_init<<<(unsigned)((NF + T - 1) / T), blk, 0, stream>>>(y, dinv, h, NF);
    edge_scatter <<<(unsigned)(((size_t)E * 32 + T - 1) / T), blk, 0, stream>>>(src, dst, dinv, y, h, E);
    bias_relu_inplace<<<(unsigned)((NF + T - 1) / T), blk, 0, stream>>>(h, b1, NF);

    // ---- layer 2 ----
    gemm_wmma_f32<<<(N + 31) / 32, blk, 0, stream>>>(h, H, W2, y, N);
    selfloop_init<<<(unsigned)((NF + T - 1) / T), blk, 0, stream>>>(y, dinv, out, NF);
    edge_scatter <<<(unsigned)(((size_t)E * 32 + T - 1) / T), blk, 0, stream>>>(src, dst, dinv, y, out, E);
    bias_add_inplace<<<(unsigned)((NF + T - 1) / T), blk, 0, stream>>>(out, b2, NF);
}